// Multi_Channel_Wiener_19859928776726
// MI455X (gfx1250) — compile-verified
//
#include <hip/hip_runtime.h>
#include <stdint.h>

// Problem constants (from reference): B=16, F=257, T=2048, C=8, N_CHUNKS=4.
// Sequence = (b,f,chunk): q = (b*F+f)*4 + chunk, 0..16447. Each sequence's
// mic data (Tc*C floats), enh data (Tc floats) and output (Tc floats) are
// fully contiguous blocks at q*Tc*C, q*Tc, q*Tc respectively.

static constexpr int kC          = 8;
static constexpr int kTc         = 512;
static constexpr int kNSeq       = 16 * 257 * 4;   // 16448
static constexpr int kSeqPerWave = 32;             // one sequence per lane (wave32)
static constexpr int kTT         = 16;             // time-tile staged per async round
static constexpr int kNTiles     = kTc / kTT;      // 32

// LDS staging buffers (per single-wave workgroup):
//  mic: [buf][t][seq-slot of 12 floats]  -> 48B slot (32B data + 16B pad, keeps
//       async b128 chunks 16B-aligned; 2-way bank conflict worst case on read)
//  enh: [buf][seq][17 floats]            -> stride 17 => conflict-free b32 reads
//  out: [buf][seq][16 floats]            -> contiguous 64B per seq for b128 async store
struct __align__(16) WaveSmem {
  float mic[2][kTT][kSeqPerWave * 12];   // 2*16*384*4  = 49152 B
  float enh[2][kSeqPerWave][17];         // 2*32*17*4   =  4352 B
  float outb[2][kSeqPerWave][kTT];       // 2*32*16*4   =  4096 B
};                                       // total ~57.6 KB of 320 KB

// ---- CDNA5 async-tensor data movement (inline asm: portable across toolchains) ----
__device__ __forceinline__ void async_load_lds_b128(uint32_t lds, uint64_t gaddr) {
  asm volatile("global_load_async_to_lds_b128 %0, %1, off" :: "v"(lds), "v"(gaddr) : "memory");
}
__device__ __forceinline__ void async_load_lds_b32(uint32_t lds, uint64_t gaddr) {
  asm volatile("global_load_async_to_lds_b32 %0, %1, off" :: "v"(lds), "v"(gaddr) : "memory");
}
__device__ __forceinline__ void async_store_lds_b128(uint64_t gaddr, uint32_t lds) {
  asm volatile("global_store_async_from_lds_b128 %0, %1, off" :: "v"(gaddr), "v"(lds) : "memory");
}
__device__ __forceinline__ void wait_async0() {
  asm volatile("s_wait_asynccnt 0" ::: "memory");
}
__device__ __forceinline__ void wait_ds0() {
  asm volatile("s_wait_dscnt 0" ::: "memory");
}

// Upper-triangle index for symmetric 8x8 (r<=j stored): off(r)=8r - r(r-1)/2
__device__ constexpr int triidx(int r, int j) {
  return (r <= j) ? (r * 8 - (r * (r - 1)) / 2 + (j - r))
                  : (j * 8 - (j * (j - 1)) / 2 + (r - j));
}

__global__ __launch_bounds__(32)
void wiener_sm_scan(const float* __restrict__ enh_g,
                    const float* __restrict__ mic_g,
                    float* __restrict__ out_g)
{
  __shared__ WaveSmem sm;

  const int      lane = threadIdx.x;          // lane == local sequence index
  const uint32_t q0   = blockIdx.x * kSeqPerWave;

  // Byte bases for this wave's 32 contiguous sequences
  const uint64_t micG = (uint64_t)(uintptr_t)mic_g + (uint64_t)q0 * (uint64_t)(kTc * kC * 4);
  const uint64_t enhG = (uint64_t)(uintptr_t)enh_g + (uint64_t)q0 * (uint64_t)(kTc * 4);
  const uint64_t outG = (uint64_t)(uintptr_t)out_g + (uint64_t)q0 * (uint64_t)(kTc * 4);

  const uint32_t micL[2] = { (uint32_t)(uintptr_t)&sm.mic[0][0][0],
                             (uint32_t)(uintptr_t)&sm.mic[1][0][0] };
  const uint32_t enhL[2] = { (uint32_t)(uintptr_t)&sm.enh[0][0][0],
                             (uint32_t)(uintptr_t)&sm.enh[1][0][0] };
  const uint32_t outL[2] = { (uint32_t)(uintptr_t)&sm.outb[0][0][0],
                             (uint32_t)(uintptr_t)&sm.outb[1][0][0] };

  // Stage one time-tile (16 steps) of mic+enh for all 32 sequences into LDS.
  // mic: issue i copies sequence i's contiguous 512B tile (lane l -> bytes l*16..+15)
  // enh: issue i covers sequences 2i,2i+1 (lane<16 -> seq 2i, lane>=16 -> seq 2i+1)
  auto stage_in = [&](int buf, int tile) {
    uint64_t msrc = micG + (uint64_t)(tile * (kTT * kC * 4)) + (uint32_t)(lane * 16);
    uint32_t mdst = micL[buf] + (uint32_t)((lane >> 1) * (kSeqPerWave * 12 * 4) + (lane & 1) * 16);
#pragma unroll 8
    for (int i = 0; i < 32; ++i) {
      async_load_lds_b128(mdst, msrc);
      mdst += 48;                              // next sequence slot in this t-row
      msrc += (uint64_t)(kTc * kC * 4);        // next sequence's global block (16 KB)
    }
    uint64_t esrc = enhG + (uint64_t)(tile * (kTT * 4))
                  + (uint32_t)((lane >> 4) * (kTc * 4) + (lane & 15) * 4);
    uint32_t edst = enhL[buf] + (uint32_t)((lane >> 4) * 68 + (lane & 15) * 4);
#pragma unroll 8
    for (int i = 0; i < 16; ++i) {
      async_load_lds_b32(edst, esrc);
      edst += 2 * 68;                          // 2 sequences per issue
      esrc += (uint64_t)(2 * kTc * 4);
    }
  };

  // Per-lane scan state: symmetric 8x8 inverse (upper triangle) + corr vector.
  float M[36];
#pragma unroll
  for (int i = 0; i < 36; ++i) M[i] = 0.0f;
#pragma unroll
  for (int r = 0; r < 8; ++r) M[triidx(r, r)] = 100.0f;   // EPS * I
  float corr[8];
#pragma unroll
  for (int j = 0; j < 8; ++j) corr[j] = 0.01f;

  const float LAM = (float)(0.99 / (1.0 - 0.99));   // lambda = 99
  const float SC  = 1.0f / (float)(1.0 - 0.99);     // 1/(1-alpha_xx)
  const float AXY = 0.01f;
  const float BXY = (float)(1.0 - 0.01);

  stage_in(0, 0);
  wait_async0();

  for (int k = 0; k < kNTiles; ++k) {
    const int cur = k & 1;
    if (k + 1 < kNTiles) stage_in(cur ^ 1, k + 1);   // prefetch next tile (async)

#pragma unroll 4
    for (int t = 0; t < kTT; ++t) {
      const float4 va = *reinterpret_cast<const float4*>(&sm.mic[cur][t][lane * 12]);
      const float4 vb = *reinterpret_cast<const float4*>(&sm.mic[cur][t][lane * 12 + 4]);
      float v[8] = { va.x, va.y, va.z, va.w, vb.x, vb.y, vb.z, vb.w };
      const float e = sm.enh[cur][lane][t];

      // mv = M v   (M symmetric, triangle storage)
      float mv[8];
#pragma unroll
      for (int r = 0; r < 8; ++r) {
        float acc = 0.0f;
#pragma unroll
        for (int j = 0; j < 8; ++j) acc = fmaf(M[triidx(r, j)], v[j], acc);
        mv[r] = acc;
      }
      // quad = v . mv  (pairwise tree: short dependence into the reciprocal)
      const float q01 = fmaf(v[1], mv[1], v[0] * mv[0]);
      const float q23 = fmaf(v[3], mv[3], v[2] * mv[2]);
      const float q45 = fmaf(v[5], mv[5], v[4] * mv[4]);
      const float q67 = fmaf(v[7], mv[7], v[6] * mv[6]);
      const float quad = (q01 + q23) + (q45 + q67);

      const float denom = fmaf(LAM, quad, 1.0f);     // 1 + lam*quad (>0: M is SPD)
      const float rden  = __builtin_amdgcn_rcpf(denom);
      const float kk    = LAM * rden;

      // M <- (M - kk * mv mv^T) * SC    (Sherman-Morrison, rank-1, symmetric)
      float tr[8];
#pragma unroll
      for (int r = 0; r < 8; ++r) tr[r] = kk * mv[r];
#pragma unroll
      for (int r = 0; r < 8; ++r) {
#pragma unroll
        for (int j = 0; j < 8; ++j) {
          if (j >= r) M[triidx(r, j)] = SC * fmaf(-tr[r], mv[j], M[triidx(r, j)]);
        }
      }

      // corr <- (1-axy)*corr + axy * v * e
      const float ae = AXY * e;
#pragma unroll
      for (int j = 0; j < 8; ++j) corr[j] = fmaf(ae, v[j], BXY * corr[j]);

      // out = (M_new corr_new) . v. Using symmetry and the SM identity:
      //   M_new v = SC * mv / denom   (since 1 - kk*quad = 1/denom exactly)
      // => out = (SC/denom) * (mv . corr_new).  Replaces the 64-FMA w-matvec.
      float mvc = 0.0f;
#pragma unroll
      for (int j = 0; j < 8; ++j) mvc = fmaf(mv[j], corr[j], mvc);
      sm.outb[cur][lane][t] = (SC * rden) * mvc;
    }

    wait_async0();   // next tile's loads landed; previous tile's stores drained
    wait_ds0();      // outb writes complete in LDS before async engine reads them

    // Async store this tile's outputs: issue i covers sequences i*8..i*8+7,
    // lane l -> seq i*8 + (l>>2), 16B chunk (l&3) of that sequence's 64B row.
    uint32_t osrc = outL[cur] + (uint32_t)((lane >> 2) * (kTT * 4) + (lane & 3) * 16);
    uint64_t odst = outG + (uint64_t)(k * (kTT * 4))
                  + (uint32_t)((lane >> 2) * (kTc * 4) + (lane & 3) * 16);
#pragma unroll
    for (int i = 0; i < 4; ++i) {
      async_store_lds_b128(odst, osrc);
      osrc += 8 * (kTT * 4);
      odst += (uint64_t)(8 * kTc * 4);
    }
  }
  wait_async0();   // drain final stores (S_ENDPGM also waits idle, belt+braces)
}

extern "C" void kernel_launch(void* const* d_in, const int* in_sizes, int n_in,
                              void* d_out, int out_size, void* d_ws, size_t ws_size,
                              hipStream_t stream) {
  (void)in_sizes; (void)n_in; (void)out_size; (void)d_ws; (void)ws_size;
  const float* enh = (const float*)d_in[0];   // spec_enhanced [16,257,2048] f32
  const float* mic = (const float*)d_in[1];   // spec_mic      [16,257,2048,8] f32
  float*       out = (float*)d_out;           // [16,257,2048] f32

  dim3 grid(kNSeq / kSeqPerWave);  // 514 single-wave workgroups
  dim3 block(32);
  wiener_sm_scan<<<grid, block, 0, stream>>>(enh, mic, out);
}